// ProteinProteinInteractionPrediction_27805618275217
// MI455X (gfx1250) — compile-verified
//
#include <hip/hip_runtime.h>
#include <stdint.h>
#include <stddef.h>

// Problem constants (match the reference).
#define DIM      20
#define NN       16384
#define LAYERS   6
#define CPAD     32            // pad 20 cols -> two 16-wide WMMA col tiles
#define KC       8             // K-split chunks per M-tile (occupancy)
#define MT       (NN / 16)     // 1024 M-tiles
#define KSTEPS   ((NN / KC) / 4)  // 512 k-steps of 4 per chunk

typedef __attribute__((ext_vector_type(2))) float v2f;
typedef __attribute__((ext_vector_type(8))) float v8f;

// ---------------------------------------------------------------------------
// x[m,d] = embed[fp[m], d]
// ---------------------------------------------------------------------------
__global__ void gather_embed_kernel(const int* __restrict__ fp,
                                    const float* __restrict__ embed,
                                    float* __restrict__ x) {
    int i = blockIdx.x * blockDim.x + threadIdx.x;
    if (i < NN * DIM) {
        int m = i / DIM;
        int d = i - m * DIM;
        x[i] = embed[(long)fp[m] * DIM + d];
    }
}

// ---------------------------------------------------------------------------
// h = relu(x @ W^T + b), written pair-packed + zero-padded to CPAD cols:
//   hpk[(m>>1)*(2*CPAD) + c*2 + (m&1)]
// so the WMMA B-operand (two consecutive K values per lane) is one b64 load.
// ---------------------------------------------------------------------------
__global__ void linear_relu_pack_kernel(const float* __restrict__ x,
                                        const float* __restrict__ W,
                                        const float* __restrict__ b,
                                        float* __restrict__ hpk) {
    int m = blockIdx.x * blockDim.x + threadIdx.x;
    if (m >= NN) return;

    float xi[DIM];
#pragma unroll
    for (int d = 0; d < DIM; ++d) xi[d] = x[(size_t)m * DIM + d];

    float* dst = hpk + (size_t)(m >> 1) * (2 * CPAD) + (m & 1);
#pragma unroll
    for (int c = 0; c < DIM; ++c) {
        float acc = b[c];
#pragma unroll
        for (int d = 0; d < DIM; ++d) acc = fmaf(xi[d], W[c * DIM + d], acc);
        dst[c * 2] = fmaxf(acc, 0.0f);
    }
#pragma unroll
    for (int c = DIM; c < CPAD; ++c) dst[c * 2] = 0.0f;  // padded cols = 0
}

// ---------------------------------------------------------------------------
// partial = adjacency-tile @ h-chunk via V_WMMA_F32_16X16X4_F32.
// One wave per (M-tile, K-chunk); computes both 16-wide column tiles so each
// adjacency element is loaded exactly once per layer.
// Grid is exact and branch-free: EXEC all-1s at every WMMA (ISA requirement).
// ---------------------------------------------------------------------------
__global__ void __launch_bounds__(256)
adj_gemm_wmma_kernel(const float* __restrict__ adj,
                     const float* __restrict__ hpk,
                     float* __restrict__ partial) {
    const int wave = (blockIdx.x * 256 + threadIdx.x) >> 5;
    const int lane = threadIdx.x & 31;
    const int kc   = wave & (KC - 1);
    const int mt   = wave / KC;

    const int m  = lane & 15;   // A row within tile / B-C-D column
    const int hi = lane >> 4;   // selects K pair {0,1} vs {2,3}
    const int k0 = kc * (NN / KC);

    // A-operand (16x4 fp32): lane holds adj[row0+m][k + 2*hi + {0,1}]
    const float* aptr = adj + (size_t)(mt * 16 + m) * NN + (k0 + 2 * hi);
    // B-operand: lane holds h[k + 2*hi + {0,1}][col] from the packed layout
    const float* bptr0 = hpk + (size_t)((k0 + 2 * hi) >> 1) * (2 * CPAD) + (m) * 2;
    const float* bptr1 = hpk + (size_t)((k0 + 2 * hi) >> 1) * (2 * CPAD) + (16 + m) * 2;

    v8f acc0 = {};
    v8f acc1 = {};

#pragma unroll 4
    for (int s = 0; s < KSTEPS; ++s) {
        // Non-temporal: adjacency is streamed (1 GiB/layer, no reuse in L2),
        // keeps the hot 2 MB hpk buffer resident.
        v2f av  = __builtin_nontemporal_load((const v2f*)aptr);
        v2f bv0 = *(const v2f*)bptr0;
        v2f bv1 = *(const v2f*)bptr1;
        acc0 = __builtin_amdgcn_wmma_f32_16x16x4_f32(
            false, av, false, bv0, (short)0, acc0, false, false);
        acc1 = __builtin_amdgcn_wmma_f32_16x16x4_f32(
            false, av, false, bv1, (short)0, acc1, false, false);
        aptr  += 4;
        bptr0 += 2 * CPAD * 2;  // advance 4 K rows in packed layout
        bptr1 += 2 * CPAD * 2;
    }

    // C/D layout: VGPR v, lane l -> D[v + 8*(l>>4)][l&15]
    float* p = partial + (size_t)(mt * KC + kc) * (2 * 256);
#pragma unroll
    for (int v = 0; v < 8; ++v) {
        p[(v + 8 * hi) * 16 + m]       = acc0[v];
        p[256 + (v + 8 * hi) * 16 + m] = acc1[v];
    }
}

// ---------------------------------------------------------------------------
// Deterministic K-chunk reduction; writes row-major [NN, DIM].
// ---------------------------------------------------------------------------
__global__ void reduce_partial_kernel(const float* __restrict__ partial,
                                      float* __restrict__ xout) {
    int i = blockIdx.x * blockDim.x + threadIdx.x;
    if (i >= NN * DIM) return;
    int mrow = i / DIM;
    int c    = i - mrow * DIM;
    int mt = mrow >> 4, r = mrow & 15;
    int ct = c >> 4,    cc = c & 15;
    const float* p = partial + (size_t)mt * KC * (2 * 256) + ct * 256 + r * 16 + cc;
    float s = 0.0f;
#pragma unroll
    for (int kc = 0; kc < KC; ++kc) s += p[(size_t)kc * (2 * 256)];
    xout[i] = s;
}

// ---------------------------------------------------------------------------
extern "C" void kernel_launch(void* const* d_in, const int* in_sizes, int n_in,
                              void* d_out, int out_size, void* d_ws, size_t ws_size,
                              hipStream_t stream) {
    const int*   fp    = (const int*)d_in[0];     // fingerprints1 [NN]
    const float* adj   = (const float*)d_in[1];   // adjacency1 [NN,NN]
    const float* embed = (const float*)d_in[2];   // [N_FINGERPRINT, DIM]
    const float* W     = (const float*)d_in[3];   // [LAYERS, DIM, DIM]
    const float* bias  = (const float*)d_in[4];   // [LAYERS, DIM]
    float*       out   = (float*)d_out;           // [NN, DIM]

    // Workspace carve-up (~20 MB total).
    float* x       = (float*)d_ws;                      // NN*DIM
    float* hpk     = x + (size_t)NN * DIM;              // (NN/2)*(2*CPAD) = NN*CPAD
    float* partial = hpk + (size_t)NN * CPAD;           // MT*KC*512

    gather_embed_kernel<<<(NN * DIM + 255) / 256, 256, 0, stream>>>(fp, embed, x);

    for (int l = 0; l < LAYERS; ++l) {
        linear_relu_pack_kernel<<<NN / 256, 256, 0, stream>>>(
            x, W + (size_t)l * DIM * DIM, bias + (size_t)l * DIM, hpk);

        adj_gemm_wmma_kernel<<<(MT * KC) / 8, 256, 0, stream>>>(adj, hpk, partial);

        reduce_partial_kernel<<<(NN * DIM + 255) / 256, 256, 0, stream>>>(
            partial, (l == LAYERS - 1) ? out : x);
    }
}